// Net_32023276159004
// MI455X (gfx1250) — compile-verified
//
#include <hip/hip_runtime.h>
#include <hip/hip_bf16.h>

typedef __attribute__((ext_vector_type(2))) float v2f;
typedef __attribute__((ext_vector_type(8))) float v8f;

#define NNODES 50000
#define NEDGES 800000
#define DIM 64

// LDS row stride (in floats) for interleaved W: 32 K-pairs x (64 cols x 2).
// 160 = 32 (mod 64) banks -> hi=0/hi=1 half-waves hit disjoint bank halves.
#define WSTRIDE 160

// ---------------------------------------------------------------------------
// Scatter-add aggregation: agg[dst[e]] += h[src[e]]  (16 threads per edge,
// each thread moves a float4 -> fully coalesced 256B per edge, 4 non-returning
// global_atomic_add_f32 per thread).
// ---------------------------------------------------------------------------
__global__ __launch_bounds__(256) void scatter_add_kernel(
    const float* __restrict__ h, const int* __restrict__ src,
    const int* __restrict__ dst, float* __restrict__ agg, int n_edges)
{
    int t = blockIdx.x * blockDim.x + threadIdx.x;
    int e = t >> 4;
    if (e >= n_edges) return;
    int c4 = (t & 15) << 2;
    const float4 val = *(const float4*)(h + (size_t)src[e] * DIM + c4);
    float* p = agg + (size_t)dst[e] * DIM + c4;
    atomicAdd(p + 0, val.x);
    atomicAdd(p + 1, val.y);
    atomicAdd(p + 2, val.z);
    atomicAdd(p + 3, val.w);
}

// ---------------------------------------------------------------------------
// out = relu(A @ W + bias), A:[N,64] f32, W:[64,64] f32.
// One wave per 16-row tile; 4 column tiles -> 4 v8f accumulators.
// K-loop: 16 steps of V_WMMA_F32_16X16X4_F32.
// A frag: lane (l,hi) holds A[m0+l][4k+2hi .. +1]            (global b64)
// B frag: W staged in LDS interleaved [K/2][n][K&1] so each fragment is one
//         contiguous ds_load_b64, bank-conflict free (stride 160).
// C/D:    VGPR r = row m0 + 8*hi + r, col = ntile*16 + l
// ---------------------------------------------------------------------------
__global__ __launch_bounds__(256) void gemm_bias_relu_kernel(
    const float* __restrict__ A, const float* __restrict__ W,
    const float* __restrict__ bias, float* __restrict__ out, int n_tiles)
{
    __shared__ float Ws[32 * WSTRIDE];
    for (int i = threadIdx.x; i < DIM * DIM; i += 256) {
        int k = i >> 6;            // 0..63
        int n = i & 63;
        Ws[(k >> 1) * WSTRIDE + n * 2 + (k & 1)] = W[i];
    }
    __syncthreads();

    int wave = (blockIdx.x * 256 + threadIdx.x) >> 5;
    if (wave >= n_tiles) return;               // wave-uniform: EXEC stays all-1
    int lane = threadIdx.x & 31;
    int l  = lane & 15;
    int hi = lane >> 4;
    int m0 = wave << 4;

    const float* arow = A + (size_t)(m0 + l) * DIM + 2 * hi;
    v8f zero = {};
    v8f acc0 = zero, acc1 = zero, acc2 = zero, acc3 = zero;

#pragma unroll
    for (int k = 0; k < 16; ++k) {
        v2f a = *(const v2f*)(arow + 4 * k);
        const float* wrow = Ws + (2 * k + hi) * WSTRIDE + l * 2;
        v2f b0 = *(const v2f*)(wrow);
        v2f b1 = *(const v2f*)(wrow + 32);
        v2f b2 = *(const v2f*)(wrow + 64);
        v2f b3 = *(const v2f*)(wrow + 96);
        acc0 = __builtin_amdgcn_wmma_f32_16x16x4_f32(false, a, false, b0, (short)0, acc0, false, false);
        acc1 = __builtin_amdgcn_wmma_f32_16x16x4_f32(false, a, false, b1, (short)0, acc1, false, false);
        acc2 = __builtin_amdgcn_wmma_f32_16x16x4_f32(false, a, false, b2, (short)0, acc2, false, false);
        acc3 = __builtin_amdgcn_wmma_f32_16x16x4_f32(false, a, false, b3, (short)0, acc3, false, false);
    }

    float bs0 = bias[l], bs1 = bias[l + 16], bs2 = bias[l + 32], bs3 = bias[l + 48];
    int mb = m0 + 8 * hi;
#pragma unroll
    for (int r = 0; r < 8; ++r) {
        float* orow = out + (size_t)(mb + r) * DIM;
        orow[l]      = fmaxf(acc0[r] + bs0, 0.0f);
        orow[l + 16] = fmaxf(acc1[r] + bs1, 0.0f);
        orow[l + 32] = fmaxf(acc2[r] + bs2, 0.0f);
        orow[l + 48] = fmaxf(acc3[r] + bs3, 0.0f);
    }
}

// ---------------------------------------------------------------------------
// Heads: xa = h @ Wa + ba  ([N,2]),  xb = h @ Wb + bb  ([N,16]).
// Wb is 64x16 -> exactly one WMMA column tile. Wa (64x2) is zero-padded to a
// 64x16 tile at staging time; only cols 0..1 are stored. Both weight sets are
// staged interleaved in LDS so each B fragment is one ds_load_b64.
// Stride 32 floats = 32 banks -> hi=0/hi=1 half-waves on disjoint bank halves.
// ---------------------------------------------------------------------------
__global__ __launch_bounds__(256) void heads_kernel(
    const float* __restrict__ h,
    const float* __restrict__ Wa, const float* __restrict__ ba,
    const float* __restrict__ Wb, const float* __restrict__ bb,
    float* __restrict__ xa, float* __restrict__ xb, int n_tiles)
{
    __shared__ float Wsb[32 * 32];
    __shared__ float Wsa[32 * 32];
    for (int i = threadIdx.x; i < 64 * 16; i += 256) {
        int k = i >> 4;            // 0..63
        int n = i & 15;
        int idx = (k >> 1) * 32 + n * 2 + (k & 1);
        Wsb[idx] = Wb[k * 16 + n];
        Wsa[idx] = (n < 2) ? Wa[k * 2 + n] : 0.0f;
    }
    __syncthreads();

    int wave = (blockIdx.x * 256 + threadIdx.x) >> 5;
    if (wave >= n_tiles) return;
    int lane = threadIdx.x & 31;
    int l  = lane & 15;
    int hi = lane >> 4;
    int m0 = wave << 4;

    const float* arow = h + (size_t)(m0 + l) * DIM + 2 * hi;
    v8f za = {}, zb = {};

#pragma unroll
    for (int k = 0; k < 16; ++k) {
        v2f a = *(const v2f*)(arow + 4 * k);
        int widx = (2 * k + hi) * 32 + l * 2;
        v2f fb = *(const v2f*)(Wsb + widx);
        v2f fa = *(const v2f*)(Wsa + widx);
        zb = __builtin_amdgcn_wmma_f32_16x16x4_f32(false, a, false, fb, (short)0, zb, false, false);
        za = __builtin_amdgcn_wmma_f32_16x16x4_f32(false, a, false, fa, (short)0, za, false, false);
    }

    float biasb = bb[l];
    float biasa = (l < 2) ? ba[l] : 0.0f;
    int mb = m0 + 8 * hi;
#pragma unroll
    for (int r = 0; r < 8; ++r) {
        xb[(size_t)(mb + r) * 16 + l] = zb[r] + biasb;
        if (l < 2) xa[(size_t)(mb + r) * 2 + l] = za[r] + biasa;
    }
}

// ---------------------------------------------------------------------------
extern "C" void kernel_launch(void* const* d_in, const int* in_sizes, int n_in,
                              void* d_out, int out_size, void* d_ws, size_t ws_size,
                              hipStream_t stream) {
    const float* v   = (const float*)d_in[0];
    const int*   src = (const int*)  d_in[1];
    const int*   dst = (const int*)  d_in[2];
    const float* W1  = (const float*)d_in[3];
    const float* b1  = (const float*)d_in[4];
    const float* W2  = (const float*)d_in[5];
    const float* b2  = (const float*)d_in[6];
    const float* Wa  = (const float*)d_in[7];
    const float* ba  = (const float*)d_in[8];
    const float* Wb  = (const float*)d_in[9];
    const float* bb  = (const float*)d_in[10];

    float* agg = (float*)d_ws;                       // [N, 64]
    float* h   = agg + (size_t)NNODES * DIM;         // [N, 64]
    float* xa  = (float*)d_out;                      // [N, 2]
    float* xb  = xa + (size_t)NNODES * 2;            // [N, 16]

    const size_t featBytes = (size_t)NNODES * DIM * sizeof(float);
    const int tiles = NNODES / 16;                   // 3125, exact
    dim3 sblk(256), sgrd((NEDGES * 16 + 255) / 256);
    dim3 gblk(256), ggrd((tiles * 32 + 255) / 256);

    // Layer 1
    hipMemsetAsync(agg, 0, featBytes, stream);
    scatter_add_kernel<<<sgrd, sblk, 0, stream>>>(v, src, dst, agg, NEDGES);
    gemm_bias_relu_kernel<<<ggrd, gblk, 0, stream>>>(agg, W1, b1, h, tiles);

    // Layer 2
    hipMemsetAsync(agg, 0, featBytes, stream);
    scatter_add_kernel<<<sgrd, sblk, 0, stream>>>(h, src, dst, agg, NEDGES);
    gemm_bias_relu_kernel<<<ggrd, gblk, 0, stream>>>(agg, W2, b2, h, tiles);

    // Heads
    heads_kernel<<<ggrd, gblk, 0, stream>>>(h, Wa, ba, Wb, bb, xa, xb, tiles);
}